// ClassificationHead_75969381532412
// MI455X (gfx1250) — compile-verified
//
#include <hip/hip_runtime.h>
#include <math.h>

// Problem sizes (fixed by the reference).
#define B_ 64
#define S_ 512
#define H_ 1024
#define L_ 30

typedef __attribute__((ext_vector_type(2))) float v2f;
typedef __attribute__((ext_vector_type(8))) float v8f;

// ---------------------------------------------------------------------------
// Kernel 1: locate marker tokens (unique by construction; default 0 matches
// jnp.argmax first-occurrence semantics) and do weighted entity pooling:
//   ent[b] = (2*f[ss] + 2*f[se] + f[os] + f[oe]) / 6
// ---------------------------------------------------------------------------
__global__ void pool_entities_kernel(const float* __restrict__ features,
                                     const int* __restrict__ sub_mask,
                                     const int* __restrict__ obj_mask,
                                     float* __restrict__ ent) {
    const int b = blockIdx.x;
    __shared__ int idx[4];  // ss, se, os, oe
    if (threadIdx.x < 4) idx[threadIdx.x] = 0;
    __syncthreads();

    const int* sm = sub_mask + b * S_;
    const int* om = obj_mask + b * S_;
    for (int s = threadIdx.x; s < S_; s += blockDim.x) {
        const int v = sm[s];
        if (v == 7)  idx[0] = s;
        if (v == 8)  idx[1] = s;
        const int w = om[s];
        if (w == 9)  idx[2] = s;
        if (w == 10) idx[3] = s;
    }
    __syncthreads();

    const float* fb  = features + (size_t)b * S_ * H_;
    const float* fss = fb + (size_t)idx[0] * H_;
    const float* fse = fb + (size_t)idx[1] * H_;
    const float* fos = fb + (size_t)idx[2] * H_;
    const float* foe = fb + (size_t)idx[3] * H_;
    float* e = ent + (size_t)b * H_;
    const float inv6 = 1.0f / 6.0f;
    for (int h = threadIdx.x; h < H_; h += blockDim.x) {
        e[h] = (2.0f * fss[h] + 2.0f * fse[h] + fos[h] + foe[h]) * inv6;
    }
}

// ---------------------------------------------------------------------------
// Kernel 2: h = gelu(ent @ W1^T + b1)  via V_WMMA_F32_16X16X4_F32.
// One wave32 per 16x16 output tile; K marched in steps of 8 with two
// independent accumulator chains to hide WMMA latency.
//
// f32 fragment layouts (ISA 7.12.2):
//   A 16x4 : lane l (row M = l&15) holds contiguous pair at K = k0 + 2*(l>>4)
//   B 4x16 : lane l (col N = l&15) holds contiguous pair at K = k0 + 2*(l>>4)
//   C/D    : VGPR r -> M = (l>>4)*8 + r, N = l&15
// ---------------------------------------------------------------------------
__global__ void gemm1_gelu_kernel(const float* __restrict__ ent,   // [B, H]
                                  const float* __restrict__ W1,    // [H, H]
                                  const float* __restrict__ bias1, // [H]
                                  float* __restrict__ hbuf) {      // [B, H]
    const int lane = threadIdx.x;          // 0..31 (wave32)
    const int n0   = blockIdx.x * 16;      // output-feature tile
    const int m0   = blockIdx.y * 16;      // batch tile
    const int mi   = lane & 15;
    const int half = lane >> 4;            // selects K sub-pair

    const float* arow = ent + (size_t)(m0 + mi) * H_;  // A: ent[m, :]
    const float* brow = W1  + (size_t)(n0 + mi) * H_;  // B: W1[n, :] (h = ent @ W1^T)

    v8f acc0 = {};
    v8f acc1 = {};
    for (int k0 = 0; k0 < H_; k0 += 8) {
        const int ka = k0 + 2 * half;
        const v2f a0 = *(const v2f*)(arow + ka);
        const v2f b0 = *(const v2f*)(brow + ka);
        const v2f a1 = *(const v2f*)(arow + ka + 4);
        const v2f b1 = *(const v2f*)(brow + ka + 4);
        acc0 = __builtin_amdgcn_wmma_f32_16x16x4_f32(false, a0, false, b0,
                                                     (short)0, acc0, false, false);
        acc1 = __builtin_amdgcn_wmma_f32_16x16x4_f32(false, a1, false, b1,
                                                     (short)0, acc1, false, false);
    }

    const int   n  = n0 + mi;
    const float bv = bias1[n];
    const int   mb = m0 + half * 8;
#pragma unroll
    for (int r = 0; r < 8; ++r) {
        const float x = acc0[r] + acc1[r] + bv;
        // exact GELU: 0.5*x*(1+erf(x/sqrt(2)))
        const float g = 0.5f * x * (1.0f + erff(x * 0.70710678118654752440f));
        hbuf[(size_t)(mb + r) * H_ + n] = g;
    }
}

// ---------------------------------------------------------------------------
// Kernel 3: out = h @ W2^T + b2.  N = 30 padded to 32; lanes mapping to
// columns >= 30 clamp their B row pointer (EXEC must stay all-1s through
// the WMMA) and skip the store.
// ---------------------------------------------------------------------------
__global__ void gemm2_kernel(const float* __restrict__ hbuf,  // [B, H]
                             const float* __restrict__ W2,    // [L, H]
                             const float* __restrict__ bias2, // [L]
                             float* __restrict__ out) {       // [B, L]
    const int lane = threadIdx.x;
    const int n0   = blockIdx.x * 16;
    const int m0   = blockIdx.y * 16;
    const int mi   = lane & 15;
    const int half = lane >> 4;

    const int  n      = n0 + mi;
    const bool nvalid = (n < L_);
    const int  nc     = nvalid ? n : 0;    // clamped row for safe loads

    const float* arow = hbuf + (size_t)(m0 + mi) * H_;
    const float* brow = W2   + (size_t)nc * H_;

    v8f acc0 = {};
    v8f acc1 = {};
    for (int k0 = 0; k0 < H_; k0 += 8) {
        const int ka = k0 + 2 * half;
        const v2f a0 = *(const v2f*)(arow + ka);
        const v2f b0 = *(const v2f*)(brow + ka);
        const v2f a1 = *(const v2f*)(arow + ka + 4);
        const v2f b1 = *(const v2f*)(brow + ka + 4);
        acc0 = __builtin_amdgcn_wmma_f32_16x16x4_f32(false, a0, false, b0,
                                                     (short)0, acc0, false, false);
        acc1 = __builtin_amdgcn_wmma_f32_16x16x4_f32(false, a1, false, b1,
                                                     (short)0, acc1, false, false);
    }

    if (nvalid) {
        const float bv = bias2[n];
        const int   mb = m0 + half * 8;
#pragma unroll
        for (int r = 0; r < 8; ++r) {
            out[(size_t)(mb + r) * L_ + n] = acc0[r] + acc1[r] + bv;
        }
    }
}

// ---------------------------------------------------------------------------
// Launch wrapper.
// Inputs (setup_inputs order): features, sub_mask, obj_mask, W1, b1, W2, b2.
// Workspace layout: ent [B*H] floats, then h [B*H] floats (512 KB total).
// ---------------------------------------------------------------------------
extern "C" void kernel_launch(void* const* d_in, const int* in_sizes, int n_in,
                              void* d_out, int out_size, void* d_ws, size_t ws_size,
                              hipStream_t stream) {
    const float* features = (const float*)d_in[0];
    const int*   sub_mask = (const int*)d_in[1];
    const int*   obj_mask = (const int*)d_in[2];
    const float* W1       = (const float*)d_in[3];
    const float* b1       = (const float*)d_in[4];
    const float* W2       = (const float*)d_in[5];
    const float* b2       = (const float*)d_in[6];
    float*       out      = (float*)d_out;

    float* ent  = (float*)d_ws;
    float* hbuf = ent + (size_t)B_ * H_;

    // 1) entity pooling
    pool_entities_kernel<<<B_, 256, 0, stream>>>(features, sub_mask, obj_mask, ent);

    // 2) h = gelu(ent @ W1^T + b1): 64 N-tiles x 4 M-tiles, one wave each
    dim3 g1(H_ / 16, B_ / 16);
    gemm1_gelu_kernel<<<g1, 32, 0, stream>>>(ent, W1, b1, hbuf);

    // 3) out = h @ W2^T + b2: 2 N-tiles (30->32) x 4 M-tiles
    dim3 g2((L_ + 15) / 16, B_ / 16);
    gemm2_kernel<<<g2, 32, 0, stream>>>(hbuf, W2, b2, out);
}